// DetectionHead_33311766347832
// MI455X (gfx1250) — compile-verified
//
#include <hip/hip_runtime.h>
#include <hip/hip_bf16.h>

// ---------------------------------------------------------------------------
// Detection head on MI455X (gfx1250, wave32, WMMA).
//   features [1,256,336,336] f32, rois [2000,4] i32
//   pool -> [2000, 12544] ; FC1(relu) -> [2000,256] ; FC2(relu) -> [2000,256]
//   heads -> cls [2000,2] ++ reg [2000,4] (concat flat in d_out)
// ---------------------------------------------------------------------------

#define POOLP 7
#define NROIS 2000
#define NCH   256
#define FH    336
#define FW    336
#define FEAT  256                      // hidden width (F)
#define K1    (NCH * POOLP * POOLP)    // 12544

typedef __attribute__((ext_vector_type(2))) float v2f;
typedef __attribute__((ext_vector_type(8))) float v8f;

// ---------------------------------------------------------------------------
// Kernel 1: ROI adaptive-avg-pool, one thread per (roi, channel, bin).
// bin edges match torch/JAX: lo_i = s + floor(i*L/7), hi_i = s + ceil((i+1)*L/7)
// ---------------------------------------------------------------------------
__global__ void roi_pool_kernel(const float* __restrict__ features,
                                const int*   __restrict__ rois,
                                float*       __restrict__ roi_feats) {
    int t = blockIdx.x * blockDim.x + threadIdx.x;
    if (t >= NROIS * NCH * POOLP * POOLP) return;
    int bin = t % 49;
    int c   = (t / 49) & (NCH - 1);
    int r   = t / (49 * NCH);

    int4 q = ((const int4*)rois)[r];         // x1, y1, x2, y2 (inclusive)
    int x1 = q.x, y1 = q.y;
    int Lx = q.z - q.x + 1;
    int Ly = q.w - q.y + 1;

    int px = bin % POOLP, py = bin / POOLP;
    int xlo = x1 + (px * Lx) / POOLP;
    int xhi = x1 + ((px + 1) * Lx + POOLP - 1) / POOLP;   // ceil
    int ylo = y1 + (py * Ly) / POOLP;
    int yhi = y1 + ((py + 1) * Ly + POOLP - 1) / POOLP;   // ceil

    const float* base = features + (size_t)c * FH * FW;
    float s = 0.0f;
    for (int y = ylo; y < yhi; ++y) {
        const float* rowp = base + (size_t)y * FW;
        for (int x = xlo; x < xhi; ++x) s += rowp[x];
    }
    float area = (float)((yhi - ylo) * (xhi - xlo));
    roi_feats[(size_t)r * K1 + c * 49 + bin] = s / area;
}

// ---------------------------------------------------------------------------
// Kernel 2/3: Out[N,256] = relu(A[N,K] @ Bw[K,256] + bias), fp32 WMMA 16x16x4.
// One wave per 16x64 output tile: one A fragment feeds 4 accumulators
// (4x A-reuse, 4 independent WMMA chains for ILP). Wave-uniform guard ->
// EXEC all ones at every WMMA.
//
// A 16x4 layout (ISA 7.12.2): lane&15 = M row; lanes 0-15 hold K=k,k+1 in
// VGPR0/1; lanes 16-31 hold K=k+2,k+3.   B 4x16: lane&15 = N col; lanes 0-15
// hold rows K=k (VGPR0), k+1 (VGPR1); lanes 16-31 hold K=k+2, k+3.
// C/D 16x16: acc[v] is row nT*16 + v + 8*(lane>=16), col base + (lane&15).
// ---------------------------------------------------------------------------
__global__ void gemm_relu_wmma(const float* __restrict__ A,
                               const float* __restrict__ Bw,
                               const float* __restrict__ bias,
                               float*       __restrict__ Out,
                               int Kdim) {
    const int wave = (blockIdx.x * blockDim.x + threadIdx.x) >> 5;
    if (wave >= (NROIS / 16) * (FEAT / 64)) return;   // 125*4=500, uniform/wave

    const int fG   = wave & 3;                        // 64-col group: 0..3
    const int nT   = wave >> 2;                       // 0..124
    const int lane = threadIdx.x & 31;
    const int half = lane >> 4;                       // 0 | 1
    const int l15  = lane & 15;

    const int row  = nT * 16 + l15;                   // A row for this lane
    const int cbase = fG * 64 + l15;                  // first of 4 cols (j*16)
    const float* aRow  = A  + (size_t)row * Kdim + (half << 1);
    const float* bBase = Bw + (size_t)(half << 1) * FEAT + cbase;

    v8f acc0 = {0.f,0.f,0.f,0.f,0.f,0.f,0.f,0.f};
    v8f acc1 = acc0, acc2 = acc0, acc3 = acc0;

    for (int kb = 0; kb < Kdim; kb += 32) {
        // stream A ahead through GL2 (global_prefetch_b8)
        __builtin_prefetch(aRow + kb + 32, 0, 1);
#pragma unroll
        for (int k = kb; k < kb + 32; k += 4) {
            v2f a = *(const v2f*)(aRow + k);          // 8B aligned (even idx)
            const float* bp = bBase + (size_t)k * FEAT;
            v2f b0, b1, b2, b3;
            b0.x = bp[0];        b0.y = bp[FEAT];
            b1.x = bp[16];       b1.y = bp[FEAT + 16];
            b2.x = bp[32];       b2.y = bp[FEAT + 32];
            b3.x = bp[48];       b3.y = bp[FEAT + 48];
            acc0 = __builtin_amdgcn_wmma_f32_16x16x4_f32(
                       false, a, false, b0, (short)0, acc0, false, false);
            acc1 = __builtin_amdgcn_wmma_f32_16x16x4_f32(
                       false, a, false, b1, (short)0, acc1, false, false);
            acc2 = __builtin_amdgcn_wmma_f32_16x16x4_f32(
                       false, a, false, b2, (short)0, acc2, false, false);
            acc3 = __builtin_amdgcn_wmma_f32_16x16x4_f32(
                       false, a, false, b3, (short)0, acc3, false, false);
        }
    }

    const int rbase = nT * 16 + half * 8;
    v8f accs[4] = {acc0, acc1, acc2, acc3};
#pragma unroll
    for (int j = 0; j < 4; ++j) {
        const int col = cbase + j * 16;
        const float bv = bias[col];
#pragma unroll
        for (int v = 0; v < 8; ++v) {
            float x = accs[j][v] + bv;
            x = x > 0.0f ? x : 0.0f;                  // ReLU
            Out[(size_t)(rbase + v) * FEAT + col] = x;
        }
    }
}

// ---------------------------------------------------------------------------
// Kernel 4: tiny heads. One thread per (roi, j), j in [0,6): j<2 -> cls, else reg.
// d_out = cls[2000*2] ++ reg[2000*4]
// ---------------------------------------------------------------------------
__global__ void heads_kernel(const float* __restrict__ h2,
                             const float* __restrict__ w_cls,
                             const float* __restrict__ b_cls,
                             const float* __restrict__ w_reg,
                             const float* __restrict__ b_reg,
                             float*       __restrict__ out) {
    int t = blockIdx.x * blockDim.x + threadIdx.x;
    if (t >= NROIS * 6) return;
    int r = t / 6, j = t % 6;
    const float* hv = h2 + (size_t)r * FEAT;
    if (j < 2) {
        float acc = b_cls[j];
        for (int k = 0; k < FEAT; ++k) acc += hv[k] * w_cls[k * 2 + j];
        out[(size_t)r * 2 + j] = acc;
    } else {
        int jj = j - 2;
        float acc = b_reg[jj];
        for (int k = 0; k < FEAT; ++k) acc += hv[k] * w_reg[k * 4 + jj];
        out[(size_t)NROIS * 2 + (size_t)r * 4 + jj] = acc;
    }
}

// ---------------------------------------------------------------------------
extern "C" void kernel_launch(void* const* d_in, const int* in_sizes, int n_in,
                              void* d_out, int out_size, void* d_ws, size_t ws_size,
                              hipStream_t stream) {
    const float* features = (const float*)d_in[0];
    const int*   rois     = (const int*)  d_in[1];
    const float* w1       = (const float*)d_in[2];
    const float* b1       = (const float*)d_in[3];
    const float* w2       = (const float*)d_in[4];
    const float* b2       = (const float*)d_in[5];
    const float* w_cls    = (const float*)d_in[6];
    const float* b_cls    = (const float*)d_in[7];
    const float* w_reg    = (const float*)d_in[8];
    const float* b_reg    = (const float*)d_in[9];
    float*       out      = (float*)d_out;

    // workspace: roi_feats [2000,12544] | h1 [2000,256] | h2 [2000,256]
    float* roi_feats = (float*)d_ws;
    float* h1        = roi_feats + (size_t)NROIS * K1;
    float* h2        = h1        + (size_t)NROIS * FEAT;

    // 1) pooling: one thread per (roi, ch, bin)
    {
        long total = (long)NROIS * NCH * POOLP * POOLP;     // 25,088,000
        int blocks = (int)((total + 255) / 256);
        roi_pool_kernel<<<blocks, 256, 0, stream>>>(features, rois, roi_feats);
    }
    // 2) FC1: 500 waves (125 nTiles x 4 col-groups), 8 waves / block
    {
        int waves  = (NROIS / 16) * (FEAT / 64);            // 500
        int blocks = (waves + 7) / 8;                       // 63
        gemm_relu_wmma<<<blocks, 256, 0, stream>>>(roi_feats, w1, b1, h1, K1);
    }
    // 3) FC2
    {
        int waves  = (NROIS / 16) * (FEAT / 64);
        int blocks = (waves + 7) / 8;
        gemm_relu_wmma<<<blocks, 256, 0, stream>>>(h1, w2, b2, h2, FEAT);
    }
    // 4) heads
    {
        int total  = NROIS * 6;
        int blocks = (total + 255) / 256;
        heads_kernel<<<blocks, 256, 0, stream>>>(h2, w_cls, b_cls, w_reg, b_reg, out);
    }
}